// FastMLNeighborSampler___9337258901923
// MI455X (gfx1250) — compile-verified
//
#include <hip/hip_runtime.h>
#include <hip/hip_bf16.h>

// ---------------------------------------------------------------------------
// Fused GraphSAGE-style neighbor sampler for MI455X (gfx1250, wave32).
//   l      = features[ids] @ W + b            (fp32 WMMA 16x16x4)
//   out    = relu(einsum('vd,vsd->vs', l, features[adj]))
//   mean / condition / sentinel masking / numnz
// One block = one 32-vertex tile (two 16-row WMMA M-tiles sharing B
// fragments in registers -> halves L2 traffic on W vs 16-row tiles).
// features table (102 MB) is L2-resident (192 MB L2), so the 512 MB
// neighbor gather is L2 bandwidth, not HBM.
// ---------------------------------------------------------------------------

typedef float v2f __attribute__((ext_vector_type(2)));
typedef float v8f __attribute__((ext_vector_type(8)));

#define NUM_NODES   100000
#define MAX_DEGREE  128
#define NODE_DIM    256
#define VERT_NUM    16384
#define NUM_SAMPLES 32
#define TILE_M      32    // vertices per block = two 16-row WMMA M-tiles
#define VF_STRIDE   260   // 256 + 4 pad: A-frag LDS reads hit distinct banks

__global__ __launch_bounds__(256) void fastml_sampler_fused(
    const int*   __restrict__ ids,
    const int*   __restrict__ unif_rand,
    const int*   __restrict__ adj_info,
    const float* __restrict__ features,
    const float* __restrict__ W,
    const float* __restrict__ bias,
    int*   __restrict__ adj_out,
    float* __restrict__ att_out,
    float* __restrict__ numnz_out,
    float* __restrict__ omean_out)
{
    __shared__ float vf[TILE_M * VF_STRIDE];            // gathered v_f tile
    __shared__ float lt[TILE_M * VF_STRIDE];            // l = v_f @ W + b tile
    __shared__ float outbuf[TILE_M * NUM_SAMPLES];      // out / notcond scratch
    __shared__ float meanbuf[TILE_M];
    __shared__ int   ids_s[TILE_M];
    __shared__ int   ur_s[NUM_SAMPLES];

    const int tid = threadIdx.x;
    const int v0  = blockIdx.x * TILE_M;

    if (tid < TILE_M)      ids_s[tid] = ids[v0 + tid];
    if (tid < NUM_SAMPLES) ur_s[tid]  = unif_rand[tid];
    __syncthreads();

    // ---- gather v_f tile into LDS: 32 rows x 256 f32, 8 threads per row ----
    {
        const int row = tid >> 3;
        const int seg = tid & 7;
        const float4* src = (const float4*)(features + (size_t)ids_s[row] * NODE_DIM) + seg * 8;
        float4*       dst = (float4*)(vf + row * VF_STRIDE) + seg * 8;
        #pragma unroll
        for (int j = 0; j < 8; ++j) dst[j] = src[j];
    }
    __syncthreads();

    // ---- phase 1: l = v_f @ W + b via V_WMMA_F32_16X16X4_F32 ----
    // A 16x4 f32 layout: lane L -> M = L&15, regs hold K = k0 + 2*(L>>4) + {0,1}
    // B 4x16  f32 layout: reg r, lane L -> K = k0 + r + 2*(L>>4), N = n0 + (L&15)
    // C/D 16x16 f32:      reg r, lane L -> M = r + 8*(L>>4),      N = n0 + (L&15)
    const int lane   = tid & 31;
    const int waveId = tid >> 5;
    const int half   = lane >> 4;
    const int lm     = lane & 15;

    const int n0a = (waveId * 2) * 16;   // 8 waves x 2 N-tiles = full N=256
    const int n0b = n0a + 16;

    v8f c00, c01, c10, c11;              // [Mtile][Ntile] accumulators
    {
        const float bv0 = bias[n0a + lm];
        const float bv1 = bias[n0b + lm];
        #pragma unroll
        for (int r = 0; r < 8; ++r) {
            c00[r] = bv0; c01[r] = bv1;
            c10[r] = bv0; c11[r] = bv1;
        }
    }

    const float* vrow0 = vf + lm * VF_STRIDE + 2 * half;              // M =  0..15
    const float* vrow1 = vf + (16 + lm) * VF_STRIDE + 2 * half;       // M = 16..31
    #pragma unroll 4
    for (int k0 = 0; k0 < NODE_DIM; k0 += 4) {
        v2f a0, a1;
        a0.x = vrow0[k0];
        a0.y = vrow0[k0 + 1];
        a1.x = vrow1[k0];
        a1.y = vrow1[k0 + 1];
        const float* wr = W + (size_t)(k0 + 2 * half) * NODE_DIM;
        v2f b0, b1;
        b0.x = wr[n0a + lm];
        b0.y = wr[NODE_DIM + n0a + lm];
        b1.x = wr[n0b + lm];
        b1.y = wr[NODE_DIM + n0b + lm];
        // each B fragment pair feeds both M-tiles: 4 WMMAs per 4 W floats
        c00 = __builtin_amdgcn_wmma_f32_16x16x4_f32(false, a0, false, b0, (short)0, c00, false, false);
        c01 = __builtin_amdgcn_wmma_f32_16x16x4_f32(false, a0, false, b1, (short)0, c01, false, false);
        c10 = __builtin_amdgcn_wmma_f32_16x16x4_f32(false, a1, false, b0, (short)0, c10, false, false);
        c11 = __builtin_amdgcn_wmma_f32_16x16x4_f32(false, a1, false, b1, (short)0, c11, false, false);
    }

    #pragma unroll
    for (int r = 0; r < 8; ++r) {
        const int m = r + 8 * half;
        lt[m * VF_STRIDE + n0a + lm]        = c00[r];
        lt[m * VF_STRIDE + n0b + lm]        = c01[r];
        lt[(16 + m) * VF_STRIDE + n0a + lm] = c10[r];
        lt[(16 + m) * VF_STRIDE + n0b + lm] = c11[r];
    }
    __syncthreads();

    // ---- phase 2: out[v,s] = relu(l[v] . features[adj[v,s]]) ----
    int   adjv[4];
    float outv[4];
    #pragma unroll
    for (int i = 0; i < 4; ++i) {
        const int p = tid + i * 256;         // 1024 (v,s) pairs per tile
        const int v = p >> 5;
        const int s = p & 31;
        const int nid = adj_info[(size_t)ids_s[v] * MAX_DEGREE + ur_s[s]];
        adjv[i] = nid;
        const float4* nf = (const float4*)(features + (size_t)nid * NODE_DIM);
        const float4* lr = (const float4*)(lt + v * VF_STRIDE);
        float acc = 0.0f;
        #pragma unroll 8
        for (int d = 0; d < NODE_DIM / 4; ++d) {
            const float4 f = nf[d];
            const float4 g = lr[d];
            acc += f.x * g.x + f.y * g.y + f.z * g.z + f.w * g.w;
        }
        const float o = fmaxf(acc, 0.0f);
        outv[i]   = o;
        outbuf[p] = o;
    }
    __syncthreads();

    // ---- row means ----
    if (tid < TILE_M) {
        float ssum = 0.0f;
        #pragma unroll
        for (int j = 0; j < NUM_SAMPLES; ++j) ssum += outbuf[tid * NUM_SAMPLES + j];
        const float m = ssum * (1.0f / NUM_SAMPLES);
        meanbuf[tid]        = m;
        omean_out[v0 + tid] = m;
    }
    __syncthreads();

    // ---- condition, masked adj, attention, notcond scratch ----
    const int sentinel = NUM_NODES - 1;
    #pragma unroll
    for (int i = 0; i < 4; ++i) {
        const int p = tid + i * 256;
        const int v = p >> 5;
        const bool cond = (adjv[i] == sentinel) || (outv[i] > 0.5f * meanbuf[v]);
        const size_t gidx = (size_t)(v0 + v) * NUM_SAMPLES + (p & 31);
        adj_out[gidx] = cond ? sentinel : adjv[i];
        att_out[gidx] = 1.0f;
        outbuf[p]     = cond ? 0.0f : 1.0f;   // each slot owned by this thread
    }
    __syncthreads();

    // ---- numnz = sum over s of notcond ----
    if (tid < TILE_M) {
        float cnt = 0.0f;
        #pragma unroll
        for (int j = 0; j < NUM_SAMPLES; ++j) cnt += outbuf[tid * NUM_SAMPLES + j];
        numnz_out[v0 + tid] = cnt;
    }
}

extern "C" void kernel_launch(void* const* d_in, const int* in_sizes, int n_in,
                              void* d_out, int out_size, void* d_ws, size_t ws_size,
                              hipStream_t stream) {
    const int*   ids  = (const int*)d_in[0];
    const int*   ur   = (const int*)d_in[1];
    const int*   adj  = (const int*)d_in[2];
    const float* feat = (const float*)d_in[3];
    const float* W    = (const float*)d_in[4];
    const float* b    = (const float*)d_in[5];

    // Outputs concatenated flat in return order:
    // adj_out (int32, 16384*32) | att_lists (f32, 16384*32) |
    // numnz (f32, 16384) | out_mean (f32, 16384)
    float* out   = (float*)d_out;
    int*   adjo  = (int*)out;                                 // int32 bits region
    float* att   = out + (size_t)VERT_NUM * NUM_SAMPLES;
    float* numnz = att + (size_t)VERT_NUM * NUM_SAMPLES;
    float* omean = numnz + VERT_NUM;

    dim3 grid(VERT_NUM / TILE_M);   // 512 tiles of 32 vertices
    fastml_sampler_fused<<<grid, 256, 0, stream>>>(ids, ur, adj, feat, W, b,
                                                   adjo, att, numnz, omean);
}